// NeuralRenderer_3178275799612
// MI455X (gfx1250) — compile-verified
//
#include <hip/hip_runtime.h>
#include <math.h>
#include <stdint.h>

typedef __attribute__((ext_vector_type(2))) float    v2f;
typedef __attribute__((ext_vector_type(8))) float    v8f;
typedef __attribute__((ext_vector_type(4))) uint32_t u32x4;
typedef __attribute__((ext_vector_type(4))) int      i32x4;
typedef __attribute__((ext_vector_type(8))) int      i32x8;

#define NFACES 640
#define NTILES (NFACES / 16)   // 40
#define IMGD   256
#define TEXD   6
#define CSTR   8               // floats per face in writeback coefficient table
#define EPSF   1e-8f
#define BIGF   1e30f

// d_ws layout:
//   [0 .. 640*8)               coef   : per-face A1,B1,C1,light,A2,B2,C2,0
//   [640*8 .. 640*8 + 40*128*2) bmat  : pre-swizzled WMMA B-matrices,
//                                        [tile][q=w0,w1,w2,z][lane] (v2f each)
#define COEF_FLOATS (NFACES * CSTR)          // 5120 floats
#define BMAT_V2F    (NTILES * 4 * 32)        // 5120 v2f = 40960 bytes

#if defined(__has_builtin)
#if __has_builtin(__builtin_amdgcn_tensor_load_to_lds) && __has_builtin(__builtin_amdgcn_s_wait_tensorcnt)
#define HAVE_TDM 1
#endif
#endif

// ---------------- per-face preprocessing ----------------
__global__ __launch_bounds__(256) void nr_prep(const float* __restrict__ verts,
                                               const int* __restrict__ faces,
                                               float* __restrict__ ws) {
  int f = blockIdx.x * blockDim.x + threadIdx.x;
  if (f >= NFACES) return;

  int ia = faces[f * 3 + 0], ib = faces[f * 3 + 1], ic = faces[f * 3 + 2];
  float ax = verts[ia*3+0], ay = verts[ia*3+1], az = verts[ia*3+2];
  float bx = verts[ib*3+0], by = verts[ib*3+1], bz = verts[ib*3+2];
  float cx = verts[ic*3+0], cy = verts[ic*3+1], cz = verts[ic*3+2];

  // face normal & lighting: light = 0.5 + 0.5 * relu(nz / (|n|+eps))
  float e1x = bx-ax, e1y = by-ay, e1z = bz-az;
  float e2x = cx-ax, e2y = cy-ay, e2z = cz-az;
  float nx = e1y*e2z - e1z*e2y;
  float ny = e1z*e2x - e1x*e2z;
  float nz = e1x*e2y - e1y*e2x;
  float nl = sqrtf(nx*nx + ny*ny + nz*nz) + EPSF;
  float light = 0.5f + 0.5f * fmaxf(nz / nl, 0.0f);

  // camera: R = identity, EYE = (0,0,-2) -> vc = v - EYE
  float z0 = az + 2.0f, z1 = bz + 2.0f, z2 = cz + 2.0f;
  float wfov = tanf(45.0f * 3.14159265358979323846f / 180.0f);
  float x0 = ax / (z0*wfov + EPSF), y0 = ay / (z0*wfov + EPSF);
  float x1 = bx / (z1*wfov + EPSF), y1 = by / (z1*wfov + EPSF);
  float x2 = cx / (z2*wfov + EPSF), y2 = cy / (z2*wfov + EPSF);

  float dx = x1-x0, dy = y1-y0;
  float ex = x2-x0, ey = y2-y0;
  float det  = dx*ey - dy*ex;
  bool  ok   = fabsf(det) > 1e-8f;
  float dets = ok ? det : 1.0f;

  float A1 =  ey/dets, B1 = -ex/dets, C1 = (y0*ex - x0*ey)/dets;
  float A2 = -dy/dets, B2 =  dx/dets, C2 = (dy*x0 - dx*y0)/dets;
  float dz1 = z1 - z0, dz2 = z2 - z0;
  float Az = A1*dz1 + A2*dz2;
  float Bz = B1*dz1 + B2*dz2;
  float Cz = z0 + C1*dz1 + C2*dz2;

  if (!ok) {   // degenerate: force w1,w2 permanently negative -> never inside
    A1 = B1 = 0.0f; C1 = -BIGF;
    A2 = B2 = 0.0f; C2 = -BIGF;
  }
  float A0 = -(A1 + A2), B0 = -(B1 + B2), C0 = 1.0f - C1 - C2;

  // writeback coefficient table (8 floats/face)
  float* c = ws + (size_t)f * CSTR;
  c[0] = A1; c[1] = B1; c[2] = C1; c[3] = light;
  c[4] = A2; c[5] = B2; c[6] = C2; c[7] = 0.0f;

  // pre-swizzled B matrices, WMMA lane layout:
  //   lanes 0-15 (K=0,1): (A,B) of face n;  lanes 16-31 (K=2,3): (C,0)
  int t = f >> 4, n = f & 15;
  v2f* row = (v2f*)(ws + COEF_FLOATS) + (size_t)t * 128;
  row[0*32 + n] = (v2f){A0, B0};  row[0*32 + n + 16] = (v2f){C0, 0.0f};
  row[1*32 + n] = (v2f){A1, B1};  row[1*32 + n + 16] = (v2f){C1, 0.0f};
  row[2*32 + n] = (v2f){A2, B2};  row[2*32 + n + 16] = (v2f){C2, 0.0f};
  row[3*32 + n] = (v2f){Az, Bz};  row[3*32 + n + 16] = (v2f){Cz, 0.0f};
}

// ---------------- rasterization via WMMA f32 16x16x4 ----------------
__global__ __launch_bounds__(256) void nr_raster(const float* __restrict__ coef,
                                                 const float* __restrict__ bmat,
                                                 const float* __restrict__ tex,
                                                 float* __restrict__ out) {
  __shared__ v2f sbm[BMAT_V2F];                    // 40 KB swizzled B table

#ifdef HAVE_TDM
  if (threadIdx.x == 0) {
    // TDM D#: 1-D tile of 10240 dwords, global bmat -> LDS sbm
    unsigned lds = (unsigned)(uintptr_t)(void*)&sbm[0];
    unsigned long long ga = (unsigned long long)(uintptr_t)bmat;
    u32x4 g0;
    g0.x = 1u;                                       // count=1 (valid), no gather
    g0.y = lds;                                      // lds_addr
    g0.z = (uint32_t)ga;                             // global_addr[31:0]
    g0.w = (uint32_t)((ga >> 32) & 0x1FFFFFFu) | (2u << 30);  // addr[56:32] | type=2
    i32x8 g1;
    g1[0] = 0x00020000;                 // wg_mask=0, data_size=2 (4 bytes)
    g1[1] = (int)(10240u << 16);        // tensor_dim0 = 10240 (bits 63:48)
    g1[2] = (int)(1u << 16);            // tensor_dim1 = 1     (bits 95:80)
    g1[3] = (int)(10240u << 16);        // tile_dim0   = 10240 (bits 127:112)
    g1[4] = 1;                          // tile_dim1   = 1     (bits 143:128)
    g1[5] = 10240;                      // tensor_dim0_stride
    g1[6] = 0; g1[7] = 0;
    i32x4 gz = {0, 0, 0, 0};
#if __clang_major__ >= 23
    i32x8 gz8 = {0, 0, 0, 0, 0, 0, 0, 0};
    __builtin_amdgcn_tensor_load_to_lds(g0, g1, gz, gz, gz8, 0);
#else
    __builtin_amdgcn_tensor_load_to_lds(g0, g1, gz, gz, 0);
#endif
    __builtin_amdgcn_s_wait_tensorcnt(0);
  }
#else
  {
    const float4* g4 = (const float4*)bmat;
    float4* s4 = (float4*)sbm;
    #pragma unroll
    for (int k = 0; k < (BMAT_V2F * 2 / 4) / 256; ++k)
      s4[k * 256 + threadIdx.x] = g4[k * 256 + threadIdx.x];
  }
#endif
  __syncthreads();

  const int lane   = threadIdx.x & 31;
  const int wave   = threadIdx.x >> 5;
  const int group  = blockIdx.x * 8 + wave;        // 16-pixel group id
  const int yrow   = group >> 4;
  const int xbase  = (group & 15) << 4;
  const int l16    = lane & 15;
  const bool hiH   = lane >= 16;

  // A matrix: rows = pixels, K = [px, py, 1, 0]
  v2f a;
  if (!hiH) {
    a.x = ((float)(xbase + l16) + 0.5f) * (2.0f / IMGD) - 1.0f;   // px (K=0)
    a.y = -(((float)yrow + 0.5f) * (2.0f / IMGD) - 1.0f);         // py (K=1)
  } else {
    a.x = 1.0f;                                                   // K=2 (const)
    a.y = 0.0f;                                                   // K=3
  }

  float bd[8]; int bi[8];
  #pragma unroll
  for (int j = 0; j < 8; ++j) { bd[j] = BIGF; bi[j] = 0; }

  #pragma unroll 2
  for (int t = 0; t < NTILES; ++t) {
    const v2f* bl = sbm + t * 128 + lane;          // conflict-free b64 loads
    v2f b0 = bl[0], b1 = bl[32], b2 = bl[64], b3 = bl[96];
    const int f = t * 16 + l16;

    v8f z = {};
    v8f W0 = __builtin_amdgcn_wmma_f32_16x16x4_f32(false, a, false, b0, (short)0, z, false, false);
    v8f W1 = __builtin_amdgcn_wmma_f32_16x16x4_f32(false, a, false, b1, (short)0, z, false, false);
    v8f W2 = __builtin_amdgcn_wmma_f32_16x16x4_f32(false, a, false, b2, (short)0, z, false, false);
    v8f DD = __builtin_amdgcn_wmma_f32_16x16x4_f32(false, a, false, b3, (short)0, z, false, false);

    #pragma unroll
    for (int j = 0; j < 8; ++j) {
      float m  = fminf(fminf(W0[j], W1[j]), W2[j]);   // v_min3: inside <=> m >= 0
      float Dj = DD[j];
      bool vis = (m >= 0.0f) && (Dj > 0.0f);
      float dm = vis ? Dj : BIGF;
      if (dm < bd[j]) { bd[j] = dm; bi[j] = f; }      // strict <: lowest face idx
    }
  }

  // argmin across the 16 faces/lanes in each half-wave (xor masks stay in half)
  #pragma unroll
  for (int j = 0; j < 8; ++j) {
    float d = bd[j]; int i = bi[j];
    #pragma unroll
    for (int m = 1; m < 16; m <<= 1) {
      float od = __shfl_xor(d, m, 32);
      int   oi = __shfl_xor(i, m, 32);
      if (od < d || (od == d && oi < i)) { d = od; i = oi; }
    }
    bd[j] = d; bi[j] = i;
  }

  // writeback: lanes 0-7 -> pixels 0..7, lanes 16-23 -> pixels 8..15
  const bool active = l16 < 8;
  const int  M   = (hiH ? 8 : 0) + l16;
  const float pxv = ((float)(xbase + M) + 0.5f) * (2.0f / IMGD) - 1.0f;
  const float pyv = -(((float)yrow + 0.5f) * (2.0f / IMGD) - 1.0f);

  #pragma unroll
  for (int j = 0; j < 8; ++j) {
    if (active && l16 == j) {
      int  fi    = bi[j];
      bool valid = bd[j] < 1e29f;
      const float4* c4 = (const float4*)(coef + (size_t)fi * CSTR);
      float4 ca = c4[0];                       // A1,B1,C1,light
      float4 cb = c4[1];                       // A2,B2,C2,0
      float W1 = ca.x * pxv + ca.y * pyv + ca.z;
      float W2 = cb.x * pxv + cb.y * pyv + cb.z;
      float W0 = 1.0f - W1 - W2;
      float light = ca.w;
      int i0 = min(max((int)(W0 * (float)TEXD), 0), TEXD - 1);
      int i1 = min(max((int)(W1 * (float)TEXD), 0), TEXD - 1);
      int i2 = min(max((int)(W2 * (float)TEXD), 0), TEXD - 1);
      const float* tp = tex + ((((size_t)fi * TEXD + i0) * TEXD + i1) * TEXD + i2) * 3;
      int p = yrow * IMGD + xbase + M;
      out[p]                   = valid ? tanhf(tp[0]) * light : 0.0f;
      out[IMGD * IMGD + p]     = valid ? tanhf(tp[1]) * light : 0.0f;
      out[2 * IMGD * IMGD + p] = valid ? tanhf(tp[2]) * light : 0.0f;
    }
  }
}

extern "C" void kernel_launch(void* const* d_in, const int* in_sizes, int n_in,
                              void* d_out, int out_size, void* d_ws, size_t ws_size,
                              hipStream_t stream) {
  const float* verts = (const float*)d_in[0];   // (1,400,3) f32
  const int*   faces = (const int*)d_in[1];     // (1,640,3) int
  const float* tex   = (const float*)d_in[2];   // (1,640,6,6,6,3) f32
  float*       out   = (float*)d_out;           // (1,3,256,256) f32
  float*       ws    = (float*)d_ws;            // 20KB coef + 40KB bmat

  nr_prep<<<(NFACES + 255) / 256, 256, 0, stream>>>(verts, faces, ws);
  nr_raster<<<(IMGD * IMGD / 16) / 8, 256, 0, stream>>>(ws, ws + COEF_FLOATS, tex, out);
}